// ModelNew_25056839205377
// MI455X (gfx1250) — compile-verified
//
#include <hip/hip_runtime.h>

typedef __attribute__((ext_vector_type(2))) float v2f;
typedef __attribute__((ext_vector_type(4))) float v4f;
typedef __attribute__((ext_vector_type(8))) float v8f;

#define NUM_N   2048
#define NUM_C   512
#define HW      49
#define SAMP    (NUM_C * HW)   // 25088 floats = 100352 B per sample
#define VEC4    (SAMP / 4)     // 6272 float4 per sample
#define BLOCK   256

__global__ __launch_bounds__(BLOCK)
void gatenorm_kernel(const float* __restrict__ x, float* __restrict__ out) {
    __shared__ alignas(16) float s_x[SAMP];   // staged sample [C][49]
    __shared__ float s_y[NUM_C];              // spatial means, then gates
    __shared__ float s_red[BLOCK];
    __shared__ float s_red2[BLOCK];
    __shared__ float s_stats[2];

    const int tid = threadIdx.x;
    const long long base = (long long)blockIdx.x * SAMP;
    const v4f* __restrict__ xin4 = reinterpret_cast<const v4f*>(x + base);
    v4f* __restrict__ out4       = reinterpret_cast<v4f*>(out + base);

    // ---- 1. Stage the whole sample into LDS (streaming: nontemporal b128) ----
    for (int i = tid; i < VEC4; i += BLOCK) {
        v4f v = __builtin_nontemporal_load(&xin4[i]);
        *reinterpret_cast<v4f*>(&s_x[4 * i]) = v;
    }
    __syncthreads();

    // ---- 2. Spatial row-sums via V_WMMA_F32_16X16X4_F32 with B == ones ----
    // A (16x4 f32) layout: lanes 0-15 hold row M=lane, K={k0,k0+1};
    //                      lanes 16-31 hold row M=lane-16, K={k0+2,k0+3}.
    // D = A * ones => D[m,n] = rowsum(m) replicated across n.
    {
        const int lane = tid & 31;
        const int wave = tid >> 5;
        const int m    = lane & 15;
        const int koff = (lane >> 4) << 1;   // 0 or 2
        v2f ones; ones[0] = 1.0f; ones[1] = 1.0f;

        for (int t = wave; t < NUM_C / 16; t += BLOCK / 32) {
            const int c0 = t * 16;
            const float* rowp = &s_x[(c0 + m) * HW];
            v8f acc = {};
            #pragma unroll
            for (int k0 = 0; k0 < 52; k0 += 4) {       // 13 steps, K padded to 52
                const int k = k0 + koff;
                v2f a;
                a[0] = (k     < HW) ? rowp[k]     : 0.0f;
                a[1] = (k + 1 < HW) ? rowp[k + 1] : 0.0f;
                acc = __builtin_amdgcn_wmma_f32_16x16x4_f32(
                        /*neg_a=*/false, a, /*neg_b=*/false, ones,
                        /*c_mod=*/(short)0, acc, /*reuse_a=*/false, /*reuse_b=*/false);
            }
            // D layout: lane with (lane&15)==0 holds M = r + 8*(lane>>4) in acc[r]
            if (m == 0) {
                const int cb = c0 + ((lane >> 4) << 3);
                #pragma unroll
                for (int r = 0; r < 8; ++r)
                    s_y[cb + r] = acc[r] * (1.0f / HW);
            }
        }
    }
    __syncthreads();

    // ---- 3. Per-sample channel stats (mean, biased var) over 512 channels ----
    {
        float a = s_y[tid], b = s_y[tid + BLOCK];
        s_red[tid]  = a + b;
        s_red2[tid] = a * a + b * b;
    }
    __syncthreads();
    for (int s = BLOCK / 2; s > 0; s >>= 1) {
        if (tid < s) {
            s_red[tid]  += s_red[tid + s];
            s_red2[tid] += s_red2[tid + s];
        }
        __syncthreads();
    }
    if (tid == 0) {
        float mean = s_red[0]  * (1.0f / NUM_C);
        float mx2  = s_red2[0] * (1.0f / NUM_C);
        float var  = fmaxf(mx2 - mean * mean, 0.0f);
        s_stats[0] = mean;
        s_stats[1] = rsqrtf(var + 1e-5f);
    }
    __syncthreads();
    const float mean    = s_stats[0];
    const float inv_std = s_stats[1];

    // ---- 4. Gate: g = exp(-0.5 * 2.0 * yn^2) = exp(-yn^2); overwrite s_y ----
    for (int c = tid; c < NUM_C; c += BLOCK) {
        float yn = (s_y[c] - mean) * inv_std;
        s_y[c] = __expf(-yn * yn);            // each thread RMWs only its own c
    }
    __syncthreads();

    // ---- 5. Gated writeback (nontemporal b128); channel = element / 49 ----
    for (int i = tid; i < VEC4; i += BLOCK) {
        const int e0 = 4 * i;
        v4f v = *reinterpret_cast<const v4f*>(&s_x[e0]);
        v4f o;
        o.x = v.x * s_y[(e0 + 0) / HW];
        o.y = v.y * s_y[(e0 + 1) / HW];
        o.z = v.z * s_y[(e0 + 2) / HW];
        o.w = v.w * s_y[(e0 + 3) / HW];
        __builtin_nontemporal_store(o, &out4[i]);
    }
}

extern "C" void kernel_launch(void* const* d_in, const int* in_sizes, int n_in,
                              void* d_out, int out_size, void* d_ws, size_t ws_size,
                              hipStream_t stream) {
    const float* x = (const float*)d_in[0];
    float* out     = (float*)d_out;
    gatenorm_kernel<<<NUM_N, BLOCK, 0, stream>>>(x, out);
}